// ConvPBC_54709293416802
// MI455X (gfx1250) — compile-verified
//
#include <hip/hip_runtime.h>

typedef __attribute__((ext_vector_type(2))) float v2f;
typedef __attribute__((ext_vector_type(8))) float v8f;

namespace {
constexpr int Bc   = 8;          // batch
constexpr int Mc   = 2;          // modes
constexpr int Lc   = 32768;      // signal length
constexpr int Kc   = 101;        // taps
constexpr int Hc   = 16;         // channels
constexpr int KP   = 104;        // taps padded to multiple of 4 (zero fill)
constexpr int LV   = Lc - Kc + 1;       // 32668 valid outputs
constexpr int HALF = Kc / 2;            // 50
constexpr int NT   = (LV + 15) / 16;    // 2042 time tiles of 16
constexpr int WAVES = 8;                // waves per block
constexpr int XST  = 128;               // staged taps per plane (>= 16+Kc+2)
}

// D = A(16x4) * B(4x16) + C, fp32 WMMA (CDNA5)
__device__ __forceinline__ v8f wmma4(v2f a, v2f b, v8f c) {
  return __builtin_amdgcn_wmma_f32_16x16x4_f32(false, a, false, b, (short)0, c,
                                               false, false);
}

// One complex-conv pass for a single mode m:
// acc[0..1]=Am(r,i) acc[2..3]=An(r,i) acc[4..5]=Ak(r,i), each a 16h x 16t tile.
__device__ __forceinline__ void conv_pass(const float* xr_l, const float* xi_l,
                                          const float* wlds, int tl, int dlt,
                                          v8f* acc) {
  for (int kb = 0; kb < Kc; kb += 4) {
    const int o = kb + dlt;
    v2f br; br.x = xr_l[tl + o]; br.y = xr_l[tl + o + 1];
    v2f bi; bi.x = xi_l[tl + o]; bi.y = xi_l[tl + o + 1];
    v2f bin; bin.x = -bi.x; bin.y = -bi.y;     // fp32 WMMA has no A/B NEG
#pragma unroll
    for (int s = 0; s < 3; ++s) {
      const float* wr = wlds + ((2 * s + 0) * Hc + tl) * KP + o;
      const float* wi = wlds + ((2 * s + 1) * Hc + tl) * KP + o;
      v2f ar; ar.x = wr[0]; ar.y = wr[1];
      v2f ai; ai.x = wi[0]; ai.y = wi[1];
      // yr = conv(xr,wr) - conv(xi,wi);  yi = conv(xr,wi) + conv(xi,wr)
      acc[2 * s + 0] = wmma4(ar, br,  acc[2 * s + 0]);
      acc[2 * s + 0] = wmma4(ai, bin, acc[2 * s + 0]);
      acc[2 * s + 1] = wmma4(ai, br,  acc[2 * s + 1]);
      acc[2 * s + 1] = wmma4(ar, bi,  acc[2 * s + 1]);
    }
  }
}

__global__ __launch_bounds__(256)
void ConvPBC_54709293416802_kernel(const float* __restrict__ x_real,
                                   const float* __restrict__ x_imag,
                                   const float* __restrict__ xpm_w,
                                   const float* __restrict__ xpm_b,
                                   const float* __restrict__ fwm_wr,
                                   const float* __restrict__ fwm_wi,
                                   const float* __restrict__ fwm_br,
                                   const float* __restrict__ fwm_bi,
                                   float* __restrict__ out) {
  __shared__ float wlds[6 * Hc * KP];          // {Am,An,Ak} x {wr,wi}, zero padded
  __shared__ float xpmw[Mc * Mc * KP];         // XPM taps, zero padded
  __shared__ float biasl[6 * Hc];              // [0,48): br  [48,96): bi
  __shared__ float xpmb[2];
  __shared__ float xstage[WAVES][2 * Mc * XST]; // per-wave {xr,xi} x {m} planes

  const int tid = threadIdx.x;

  // ---- cooperative constant staging ----
  for (int idx = tid; idx < 6 * Hc * KP; idx += 256) {
    int pl = idx / (Hc * KP);
    int rem = idx - pl * (Hc * KP);
    int h = rem / KP, k = rem - h * KP;
    int s = pl >> 1;
    const float* src = (pl & 1) ? fwm_wi : fwm_wr;
    wlds[idx] = (k < Kc) ? src[(s * Hc + h) * Kc + k] : 0.f;
  }
  for (int idx = tid; idx < Mc * Mc * KP; idx += 256) {
    int mm = idx / KP, k = idx - mm * KP;
    xpmw[idx] = (k < Kc) ? xpm_w[mm * Kc + k] : 0.f;
  }
  if (tid < 48) biasl[tid] = fwm_br[tid];
  else if (tid < 96) biasl[tid] = fwm_bi[tid - 48];
  if (tid < 2) xpmb[tid] = xpm_b[tid];
  __syncthreads();

  const int wave = tid >> 5, lane = tid & 31;
  const int unit = blockIdx.x * WAVES + wave;
  if (unit >= Bc * NT) return;                  // grid fits exactly; wave-uniform
  const int b = unit / NT;
  int t0 = (unit - b * NT) * 16;
  if (t0 > LV - 16) t0 = LV - 16;               // last tile: overlapped recompute

  const int tl  = lane & 15;                    // col t (and row h for A frags)
  const int dlt = (lane >> 4) << 1;             // +2 k-offset for upper half-wave
  const int hi8 = (lane >> 4) << 3;             // +8 row offset for C/D tiles

  float* xs = xstage[wave];
  // ---- stage this tile's signal window (coalesced, clamped) ----
  for (int m = 0; m < Mc; ++m) {
    const float* gr = x_real + (size_t)(b * Mc + m) * Lc;
    const float* gi = x_imag + (size_t)(b * Mc + m) * Lc;
#pragma unroll
    for (int c = 0; c < 4; ++c) {
      int j = lane * 4 + c;
      int g = t0 + j;
      if (g > Lc - 1) g = Lc - 1;               // pad taps feed zero weights
      xs[(m * 2 + 0) * XST + j] = gr[g];
      xs[(m * 2 + 1) * XST + j] = gi[g];
    }
  }
  __builtin_amdgcn_wave_barrier();

  // snapshot center samples x[t0+tl+HALF]; lanes>=16 hold m=1
  const int msel = lane >> 4;
  const float xcr = xs[(msel * 2 + 0) * XST + tl + HALF];
  const float xci = xs[(msel * 2 + 1) * XST + tl + HALF];

  const v8f vzero = {0.f, 0.f, 0.f, 0.f, 0.f, 0.f, 0.f, 0.f};
  v8f Sr = vzero, Si = vzero;
  v8f An0r = vzero, An0i = vzero, An1r = vzero, An1i = vzero;

  // ---- the 20-GFLOP core: 624 fp32 WMMAs per tile ----
  for (int m = 0; m < Mc; ++m) {
    v8f acc[6];
#pragma unroll
    for (int i = 0; i < 6; ++i) acc[i] = vzero;
    conv_pass(xs + (m * 2 + 0) * XST, xs + (m * 2 + 1) * XST, wlds, tl, dlt, acc);
    // add complex biases (row h = hi8 + v)
#pragma unroll
    for (int s = 0; s < 3; ++s)
#pragma unroll
      for (int v = 0; v < 8; ++v) {
        acc[2 * s + 0][v] += biasl[s * Hc + hi8 + v];
        acc[2 * s + 1][v] += biasl[48 + s * Hc + hi8 + v];
      }
    // S += Am * conj(Ak), elementwise in C/D layout
#pragma unroll
    for (int v = 0; v < 8; ++v) {
      Sr[v] += acc[0][v] * acc[4][v] + acc[1][v] * acc[5][v];
      Si[v] += acc[1][v] * acc[4][v] - acc[0][v] * acc[5][v];
    }
    if (m == 0) { An0r = acc[2]; An0i = acc[3]; }
    else        { An1r = acc[2]; An1i = acc[3]; }
  }

  // ---- XPM phase: overwrite xr planes with |x|^2, then tiny conv ----
  __builtin_amdgcn_wave_barrier();
#pragma unroll
  for (int m = 0; m < Mc; ++m)
    for (int j = lane; j < XST; j += 32) {
      float a = xs[(m * 2 + 0) * XST + j];
      float c = xs[(m * 2 + 1) * XST + j];
      xs[(m * 2 + 0) * XST + j] = a * a + c * c;
    }
  __builtin_amdgcn_wave_barrier();

  const int mo = lane >> 4;                     // lanes<16: m=0, lanes>=16: m=1
  float phi = xpmb[mo];
#pragma unroll
  for (int mi = 0; mi < Mc; ++mi) {
    const float* pp = xs + (mi * 2) * XST + tl;
    const float* ww = xpmw + (mo * Mc + mi) * KP;
#pragma unroll 4
    for (int k = 0; k < Kc; ++k) phi += pp[k] * ww[k];
  }
  float sphi, cphi;
  __sincosf(phi, &sphi, &cphi);
  // pull m=1 per-t values down to lanes 0..15
  const float sphi_o = __shfl_xor(sphi, 16, 32);
  const float cphi_o = __shfl_xor(cphi, 16, 32);
  const float xcr_o  = __shfl_xor(xcr, 16, 32);
  const float xci_o  = __shfl_xor(xci, 16, 32);

  float2* outc = (float2*)out;                  // [B][M][LV] x (re,im)
#pragma unroll
  for (int m = 0; m < Mc; ++m) {
    const v8f Ar = m ? An1r : An0r;
    const v8f Ai = m ? An1i : An0i;
    float pr = 0.f, pi = 0.f;
#pragma unroll
    for (int v = 0; v < 8; ++v) {               // partial sum over 8 rows h
      pr += Ar[v] * Sr[v] - Ai[v] * Si[v];
      pi += Ar[v] * Si[v] + Ai[v] * Sr[v];
    }
    pr += __shfl_xor(pr, 16, 32);               // fold h=8..15 half
    pi += __shfl_xor(pi, 16, 32);
    const float cs  = m ? cphi_o : cphi;
    const float sn  = m ? sphi_o : sphi;
    const float xr0 = m ? xcr_o : xcr;
    const float xi0 = m ? xci_o : xci;
    if (lane < 16) {
      float2 e;
      e.x = xr0 * cs - xi0 * sn + pr;
      e.y = xr0 * sn + xi0 * cs + pi;
      outc[(size_t)(b * Mc + m) * LV + t0 + tl] = e;
    }
  }
}

extern "C" void kernel_launch(void* const* d_in, const int* in_sizes, int n_in,
                              void* d_out, int out_size, void* d_ws, size_t ws_size,
                              hipStream_t stream) {
  (void)in_sizes; (void)n_in; (void)out_size; (void)d_ws; (void)ws_size;
  const float* x_real = (const float*)d_in[0];
  const float* x_imag = (const float*)d_in[1];
  const float* xpm_w  = (const float*)d_in[2];
  const float* xpm_b  = (const float*)d_in[3];
  const float* fwm_wr = (const float*)d_in[4];
  const float* fwm_wi = (const float*)d_in[5];
  const float* fwm_br = (const float*)d_in[6];
  const float* fwm_bi = (const float*)d_in[7];
  float* out = (float*)d_out;

  const int units  = Bc * NT;                   // 16336 wave-tiles
  const int blocks = (units + WAVES - 1) / WAVES;
  ConvPBC_54709293416802_kernel<<<blocks, 256, 0, stream>>>(
      x_real, x_imag, xpm_w, xpm_b, fwm_wr, fwm_wi, fwm_br, fwm_bi, out);
}